// Diffusion1D_71167608094899
// MI455X (gfx1250) — compile-verified
//
#include <hip/hip_runtime.h>
#include <math.h>

// ---------------- configuration ----------------
#define NT    256                       // threads per block (8 waves, wave32)
#define CH    64                        // elements per thread
#define TILE  (NT * CH)                 // 16384 elements per workgroup tile
#define HMAX  256                       // max halo (runtime Hn <= HMAX)
#define LSTR  66                        // padded LDS stride per 64-elem block (66 % 64 == 2 -> conflict-free)
#define NBLK_D   ((TILE + 2 * HMAX) / 64)   // 264 blocks for d / x staging
#define NBLK_DP  ((TILE + HMAX) / 64)       // 260 blocks for d'
#define SZ_D   (NBLK_D * LSTR)          // 17424 floats
#define SZ_DP  (NBLK_DP * LSTR)         // 17160 floats
#define SMEM_FLOATS (SZ_D + SZ_DP + HMAX)   // + 256 float c' table  (~139.4 KB)

static_assert((TILE + 2 * HMAX) % (2 * NT) == 0, "load loop must be exact");
static_assert(TILE % (2 * NT) == 0, "store loop must be exact");

__device__ __forceinline__ int lidx(int u) { return ((u >> 6) * LSTR) + (u & 63); }

typedef int v2i __attribute__((ext_vector_type(2)));

#if defined(__gfx1250__) && __has_builtin(__builtin_amdgcn_global_load_async_to_lds_b64)
#define HAVE_ASYNC_LOAD 1
#else
#define HAVE_ASYNC_LOAD 0
#endif
#if defined(__gfx1250__) && __has_builtin(__builtin_amdgcn_global_store_async_from_lds_b64)
#define HAVE_ASYNC_STORE 1
#else
#define HAVE_ASYNC_STORE 0
#endif

__device__ __forceinline__ void wait_async_zero() {
#if __has_builtin(__builtin_amdgcn_s_wait_asynccnt)
  __builtin_amdgcn_s_wait_asynccnt(0);
#else
  asm volatile("s_wait_asynccnt 0" ::: "memory");
#endif
}

__global__ __launch_bounds__(NT) void diffusion1d_thomas(
    const float* __restrict__ C, const float* __restrict__ p_dt,
    const float* __restrict__ p_csurf, const float* __restrict__ p_cbulk,
    const float* __restrict__ p_D, const float* __restrict__ p_dx,
    float* __restrict__ out, int n)
{
  extern __shared__ float smem[];
  float* s_d  = smem;                   // staged d (later reused to stage x): rows [S-HMAX, S+TILE+HMAX)
  float* s_dp = smem + SZ_D;            // d' : rows [S, S+TILE+HMAX)
  float* s_ct = smem + SZ_D + SZ_DP;    // exact c' for global rows [0, HMAX)

  const int tid = threadIdx.x;
  const int S   = (int)blockIdx.x * TILE;

  // runtime scalars
  const float dt = *p_dt, Cs = *p_csurf, Cb = *p_cbulk, Dc = *p_D, dxv = *p_dx;
  const float r  = Dc * dt / (dxv * dxv);
  const float b  = 1.0f + 2.0f * r;
  const float denomstar = 0.5f * (b + sqrtf(fmaf(4.0f, r, 1.0f))); // converged Thomas denom
  const float invd  = 1.0f / denomstar;
  const float beta  = r * invd;         // |c*| : per-step decay of both sweeps
  const float cstar = -beta;            // converged c'
  int Hn = 8;                            // runtime halo: beta^Hn < 2^-34
  if (beta > 1e-7f) {
    float l2 = -log2f(beta);
    Hn = (int)ceilf(34.0f / l2);
    Hn = Hn < 8 ? 8 : (Hn > HMAX ? HMAX : Hn);
  }

  // ---------- stage tile (+ halos) into LDS via async DMA ----------
  {
    const int gbase = S - HMAX;
    for (int k = 0; k < (TILE + 2 * HMAX) / (2 * NT); ++k) {   // 33 iterations
      const int e  = 2 * (k * NT + tid);
      const int g  = gbase + e;
      const int li = lidx(e);                                   // even -> 8B aligned
      if (g >= 0 && g + 1 < n) {
#if HAVE_ASYNC_LOAD
        __builtin_amdgcn_global_load_async_to_lds_b64(
            (__attribute__((address_space(1))) v2i*)(C + g),
            (__attribute__((address_space(3))) v2i*)(s_d + li), 0, 0);
#else
        *(float2*)(s_d + li) = *(const float2*)(C + g);
#endif
      } else {
        s_d[li]     = (g     >= 0 && g     < n) ? C[g]     : 0.0f;
        s_d[li + 1] = (g + 1 >= 0 && g + 1 < n) ? C[g + 1] : 0.0f;
      }
    }
  }

  // exact c' table for rows near the x=0 Dirichlet boundary (only block 0 reads it)
  if (S == 0 && tid == 0) {
    float cp = 0.0f;                    // row 0: b=1, c=0  ->  c'_0 = 0
    s_ct[0] = 0.0f;
    for (int i = 1; i < HMAX; ++i) {
      float den = fmaf(r, cp, b);       // b - a*c'_{i-1},  a = -r
      cp = -r / den;
      s_ct[i] = cp;
    }
  }

#if HAVE_ASYNC_LOAD
  wait_async_zero();
#endif
  __syncthreads();

  const int s = S + tid * CH;
  const bool active = (s < n);

  // ---------- forward sweep (per-thread, halo warm-up) ----------
  if (active) {
    const int e         = min(s + CH, n);
    const int ext       = (tid == NT - 1) ? Hn : 0;     // last thread extends d' into right halo
    const int store_end = min(e + ext, n);
    int g0 = s - Hn; if (g0 < 0) g0 = 0;
    float cp = 0.0f, dpv = 0.0f;

    if (g0 < HMAX) {
      // near global row 0: exact recurrence (matches reference row-by-row)
      int g = g0;
      if (g == 0) {
        cp = 0.0f; dpv = Cs;            // row 0: c'=0, d'=C_surf
        if (g >= s) s_dp[lidx(g - S)] = dpv;
        g = 1;
      }
      for (; g < store_end; ++g) {
        float dg  = s_d[lidx(g - S + HMAX)];
        float den = fmaf(r, cp, b);
        cp  = -r / den;
        dpv = (dg + r * dpv) / den;
        if (g >= s) s_dp[lidx(g - S)] = dpv;
      }
    } else {
      // interior: c' fully converged -> d'_i = beta*d'_{i-1} + d_i/denom*  (1 FMA chain)
      int g = g0;
      for (; g < s; ++g)                                   // warm-up, discard
        dpv = fmaf(beta, dpv, s_d[lidx(g - S + HMAX)] * invd);
      const int m_end = min(store_end, n - 1);
      for (; g < m_end; ++g) {                             // stored portion
        dpv = fmaf(beta, dpv, s_d[lidx(g - S + HMAX)] * invd);
        s_dp[lidx(g - S)] = dpv;
      }
      if (g == n - 1 && g < store_end)                     // row n-1: a=0,b=1 -> d' = C_bulk
        s_dp[lidx(g - S)] = Cb;
    }
  }
  __syncthreads();

  // ---------- backward sweep (per-thread, halo warm-up); x staged into s_d ----------
  if (active) {
    const int e = min(s + CH, n);
    int j0 = e - 1 + Hn; if (j0 > n - 1) j0 = n - 1;
    float x = s_dp[lidx(j0 - S)];          // truncated start (exact when j0 == n-1)
    if (j0 < e) s_d[lidx(j0 - S + HMAX)] = x;
    int j = j0 - 1;
    if (s >= HMAX) {
      for (; j >= e; --j)                                   // warm-up: x = d' - c*·x = d' + beta·x
        x = fmaf(beta, x, s_dp[lidx(j - S)]);
      for (; j >= s; --j) {
        x = fmaf(beta, x, s_dp[lidx(j - S)]);
        s_d[lidx(j - S + HMAX)] = x;
      }
    } else {
      for (; j >= e; --j) {
        float cj = (j < HMAX) ? s_ct[j] : cstar;
        x = s_dp[lidx(j - S)] - cj * x;
      }
      for (; j >= s; --j) {
        float cj = (j < HMAX) ? s_ct[j] : cstar;
        x = s_dp[lidx(j - S)] - cj * x;
        s_d[lidx(j - S + HMAX)] = x;
      }
    }
  }
  __syncthreads();

  // ---------- write-back of x: async LDS -> global DMA ----------
  for (int k = 0; k < TILE / (2 * NT); ++k) {               // 32 iterations
    const int eo = 2 * (k * NT + tid);
    const int g  = S + eo;
    const int li = lidx(eo + HMAX);
    if (g + 1 < n) {
#if HAVE_ASYNC_STORE
      __builtin_amdgcn_global_store_async_from_lds_b64(
          (__attribute__((address_space(1))) v2i*)(out + g),
          (__attribute__((address_space(3))) v2i*)(s_d + li), 0, 0);
#else
      float2 v; v.x = s_d[li]; v.y = s_d[li + 1];
      *(float2*)(out + g) = v;
#endif
    } else if (g < n) {
      out[g] = s_d[li];
    }
  }
  // S_ENDPGM performs an implicit wait-idle (all counters, incl. ASYNCcnt),
  // so outstanding async stores complete before the kernel retires.
}

extern "C" void kernel_launch(void* const* d_in, const int* in_sizes, int n_in,
                              void* d_out, int out_size, void* d_ws, size_t ws_size,
                              hipStream_t stream) {
  const float* C    = (const float*)d_in[0];
  const float* dt   = (const float*)d_in[1];
  const float* csur = (const float*)d_in[2];
  const float* cbul = (const float*)d_in[3];
  const float* D    = (const float*)d_in[4];
  const float* dx   = (const float*)d_in[5];
  float* out = (float*)d_out;
  const int n = in_sizes[0];
  const int blocks = (n + TILE - 1) / TILE;
  const size_t shmem = (size_t)SMEM_FLOATS * sizeof(float);
  diffusion1d_thomas<<<blocks, NT, shmem, stream>>>(C, dt, csur, cbul, D, dx, out, n);
  (void)d_ws; (void)ws_size; (void)n_in; (void)out_size;
}